// FuserPipeline_61168924230179
// MI455X (gfx1250) — compile-verified
//
#include <hip/hip_runtime.h>
#include <math.h>

// Problem constants (match reference setup_inputs)
#define NB 4
#define NC 1024
#define NN 16384
#define NBIN13 169
#define NBIN26 676
#define NBIN52 2704
#define BINS_PER_C (NBIN13 + NBIN26 + NBIN52)   // 3549

// Tiling
#define T        8        // channels per workgroup
#define CHUNK    512      // points staged per chunk
#define THREADS  256      // 8 waves (wave32)
#define NCHUNKS  (NN / CHUNK)

typedef int v4i_t __attribute__((vector_size(16)));
typedef __attribute__((address_space(1))) v4i_t* g4_t;   // global int4*
typedef __attribute__((address_space(3))) v4i_t* l4_t;   // LDS int4*

__device__ __forceinline__ void async_copy_b128(const float* g, float* l) {
    // global -> LDS async (16 bytes per lane), tracked by ASYNCcnt
    __builtin_amdgcn_global_load_async_to_lds_b128(
        (g4_t)(uintptr_t)g,
        (l4_t)(uint32_t)(uintptr_t)l,
        0 /*offset*/, 0 /*cpol*/);
}

__global__ __launch_bounds__(THREADS)
void proj_grids_kernel(const float* __restrict__ feats,
                       const int*   __restrict__ idx13,
                       const int*   __restrict__ idx26,
                       const int*   __restrict__ idx52,
                       float*       __restrict__ out)
{
    // LDS: bins for 8 channels x all three grids, shared counts, double feature buffer
    __shared__ float sbins[T * BINS_PER_C];      // 113,568 B
    __shared__ float scnt[BINS_PER_C];           //  14,196 B
    __shared__ float sfeat[2][T * CHUNK];        //  32,768 B   (total ~160 KB)

    const int tid    = threadIdx.x;
    const int ctiles = NC / T;                   // 128
    const int b      = blockIdx.x / ctiles;
    const int c0     = (blockIdx.x % ctiles) * T;

    float* bins13 = sbins;
    float* bins26 = sbins + T * NBIN13;
    float* bins52 = sbins + T * (NBIN13 + NBIN26);
    float* cnt13  = scnt;
    float* cnt26  = scnt + NBIN13;
    float* cnt52  = scnt + NBIN13 + NBIN26;

    // zero accumulators
    for (int j = tid; j < T * BINS_PER_C; j += THREADS) sbins[j] = 0.0f;
    for (int j = tid; j < BINS_PER_C;     j += THREADS) scnt[j]  = 0.0f;

    const float* gbase = feats + ((size_t)b * NC + c0) * NN;
    const int*   i13   = idx13 + (size_t)b * NN;
    const int*   i26   = idx26 + (size_t)b * NN;
    const int*   i52   = idx52 + (size_t)b * NN;

    // -------- prefetch helper: stage one chunk (T x CHUNK floats) into sfeat[buf] --------
    // (T*CHUNK)/4 float4's spread over THREADS lanes -> 4 async b128 per thread
    auto prefetch = [&](int chunk, int buf) {
        const int base = chunk * CHUNK;
#pragma unroll
        for (int i = 0; i < (T * CHUNK) / (4 * THREADS); ++i) {   // 4 iterations
            const int q  = tid + i * THREADS;        // float4 index in tile
            const int t  = q >> 7;                   // / (CHUNK/4 = 128)
            const int p4 = (q & 127) << 2;
            const float* g = gbase + (size_t)t * NN + base + p4;
            float*       l = &sfeat[buf][t * CHUNK + p4];
            async_copy_b128(g, l);
        }
    };

    prefetch(0, 0);

    for (int k = 0; k < NCHUNKS; ++k) {
        const int buf = k & 1;

        // (A) everyone finished consuming the buffer we are about to overwrite
        __syncthreads();

        if (k + 1 < NCHUNKS) {
            prefetch(k + 1, buf ^ 1);                       // 4 new async ops in flight
            __builtin_amdgcn_s_wait_asynccnt(4);            // chunk k's 4 ops are done
        } else {
            __builtin_amdgcn_s_wait_asynccnt(0);
        }
        // (B) chunk k data from ALL waves is visible in LDS
        __syncthreads();

        const int base = k * CHUNK;
#pragma unroll
        for (int pp = 0; pp < CHUNK / THREADS; ++pp) {      // 2 points per thread
            const int p = tid + pp * THREADS;
            const int n = base + p;
            const int j13 = i13[n];
            const int j26 = i26[n];
            const int j52 = i52[n];
            atomicAdd(&cnt13[j13], 1.0f);
            atomicAdd(&cnt26[j26], 1.0f);
            atomicAdd(&cnt52[j52], 1.0f);
#pragma unroll
            for (int t = 0; t < T; ++t) {
                const float f = sfeat[buf][t * CHUNK + p];
                atomicAdd(&bins13[t * NBIN13 + j13], f);
                atomicAdd(&bins26[t * NBIN26 + j26], f);
                atomicAdd(&bins52[t * NBIN52 + j52], f);
            }
        }
    }
    __syncthreads();

    // -------- post-process: avg, clamp, zero-fill, log1p; coalesced stores --------
    const size_t off26 = (size_t)NB * NC * NBIN13;
    const size_t off52 = off26 + (size_t)NB * NC * NBIN26;

    for (int j = tid; j < T * NBIN13; j += THREADS) {
        const int t = j / NBIN13, bin = j % NBIN13;
        float v = bins13[j] / (cnt13[bin] + 1e-6f);
        v = fmaxf(v, 0.0f);
        v = (v == 0.0f) ? 1e-5f : v;
        out[((size_t)(b * NC + c0 + t)) * NBIN13 + bin] = log1pf(v);
    }
    for (int j = tid; j < T * NBIN26; j += THREADS) {
        const int t = j / NBIN26, bin = j % NBIN26;
        float v = bins26[j] / (cnt26[bin] + 1e-6f);
        v = fmaxf(v, 0.0f);
        v = (v == 0.0f) ? 1e-5f : v;
        out[off26 + ((size_t)(b * NC + c0 + t)) * NBIN26 + bin] = log1pf(v);
    }
    for (int j = tid; j < T * NBIN52; j += THREADS) {
        const int t = j / NBIN52, bin = j % NBIN52;
        float v = bins52[j] / (cnt52[bin] + 1e-6f);
        v = fmaxf(v, 0.0f);
        v = (v == 0.0f) ? 1e-5f : v;
        out[off52 + ((size_t)(b * NC + c0 + t)) * NBIN52 + bin] = log1pf(v);
    }
}

extern "C" void kernel_launch(void* const* d_in, const int* in_sizes, int n_in,
                              void* d_out, int out_size, void* d_ws, size_t ws_size,
                              hipStream_t stream) {
    const float* feats = (const float*)d_in[0];
    const int*   idx13 = (const int*)d_in[1];
    const int*   idx26 = (const int*)d_in[2];
    const int*   idx52 = (const int*)d_in[3];
    float*       out   = (float*)d_out;

    const int nblocks = NB * (NC / T);   // 512 workgroups
    proj_grids_kernel<<<nblocks, THREADS, 0, stream>>>(feats, idx13, idx26, idx52, out);
}